// PermLocalSearch_14199161880839
// MI455X (gfx1250) — compile-verified
//
#include <hip/hip_runtime.h>
#include <math.h>

#define KPERM 16
#define E     128
#define PITCH 132          // rows 16B-aligned (528 B); row-pass lane stride 4 mod 64 banks (2-way)
#define EPS_  1e-6f

// ---- CDNA5 async global<->LDS ops (gfx1250), via inline asm ---------------
__device__ __forceinline__ uint32_t lds_addr_of(const void* p) {
  // low 32 bits of a flat shared pointer == LDS byte address
  return (uint32_t)(uintptr_t)p;
}
__device__ __forceinline__ void async_load_b128(uint32_t lds_byte, const void* gaddr) {
  asm volatile("global_load_async_to_lds_b128 %0, %1, off"
               :: "v"(lds_byte), "v"(gaddr) : "memory");
}
__device__ __forceinline__ void async_store_b128(void* gaddr, uint32_t lds_byte) {
  asm volatile("global_store_async_from_lds_b128 %0, %1, off"
               :: "v"(gaddr), "v"(lds_byte) : "memory");
}
__device__ __forceinline__ void wait_async0() {
  asm volatile("s_wait_asynccnt 0x0" ::: "memory");
}

// ---------------------------------------------------------------------------
// One workgroup per k-matrix. 128 threads (4 wave32). Whole 128x128 matrix
// lives in LDS (pitch 132) for all 20 Sinkhorn iterations; each pass caches
// its owned row/column in VGPRs (single LDS read sweep per pass).
// ---------------------------------------------------------------------------
__global__ __launch_bounds__(128)
void sinkhorn_entropy_kernel(const float* __restrict__ plw,
                             float* __restrict__ out_pw,
                             float* __restrict__ ws_partial,
                             float inv_tau) {
  extern __shared__ __align__(16) float smem[];
  float* mainm  = smem;              // PITCH*E floats
  float* rowLse = smem + PITCH * E;  // E floats
  float* red    = rowLse + E;        // E floats

  const int tid = threadIdx.x;
  const int k   = blockIdx.x;
  const float* src   = plw    + (size_t)k * E * E;
  float*       dstPw = out_pw + (size_t)k * E * E;
  const uint32_t lbase = lds_addr_of(mainm);

  // ---- async copy global (pitch 128) -> LDS (pitch 132), b128 chunks ------
  #pragma unroll
  for (int i = 0; i < 32; ++i) {
    int v  = tid + i * 128;          // vec4 index 0..4095
    int r  = v >> 5;                 // 32 vec4 per row
    int c4 = (v & 31) * 4;           // starting column (multiple of 4)
    async_load_b128(lbase + (uint32_t)(r * PITCH + c4) * 4u, src + (r * E + c4));
  }
  wait_async0();
  __syncthreads();

  // ---- scale by 1/tau in place (lane-consecutive, conflict-free) ----------
  for (int r = 0; r < E; ++r) mainm[r * PITCH + tid] *= inv_tau;
  __syncthreads();

  // ---- 20 Sinkhorn iterations: row lse, then column lse (row lse fused) ---
  for (int it = 0; it < 20; ++it) {
    // row logsumexp: thread tid owns row tid; one float4 LDS sweep -> regs
    {
      const float4* rp = (const float4*)(mainm + tid * PITCH);
      float4 rv[32];
      #pragma unroll
      for (int i = 0; i < 32; ++i) rv[i] = rp[i];
      float m = -INFINITY;
      #pragma unroll
      for (int i = 0; i < 32; ++i)
        m = fmaxf(m, fmaxf(fmaxf(rv[i].x, rv[i].y), fmaxf(rv[i].z, rv[i].w)));
      float s = 0.f;
      #pragma unroll
      for (int i = 0; i < 32; ++i)
        s += expf(rv[i].x - m) + expf(rv[i].y - m)
           + expf(rv[i].z - m) + expf(rv[i].w - m);
      rowLse[tid] = m + logf(s);
    }
    __syncthreads();

    // column pass: thread tid owns column tid; one read sweep -> regs,
    // subtract rowLse on the fly, one write sweep back.
    {
      float cv[E];
      #pragma unroll
      for (int r = 0; r < E; ++r)
        cv[r] = mainm[r * PITCH + tid] - rowLse[r];
      float m2 = -INFINITY;
      #pragma unroll
      for (int r = 0; r < E; ++r) m2 = fmaxf(m2, cv[r]);
      float s2 = 0.f;
      #pragma unroll
      for (int r = 0; r < E; ++r) s2 += expf(cv[r] - m2);
      const float cl = m2 + logf(s2);
      #pragma unroll
      for (int r = 0; r < E; ++r)
        mainm[r * PITCH + tid] = cv[r] - cl;
    }
    __syncthreads();
  }

  // ---- pw = exp(la); entropy terms; overwrite LDS with pw -----------------
  {
    float4* rp4 = (float4*)(mainm + tid * PITCH);
    float4 pv[32];
    #pragma unroll
    for (int i = 0; i < 32; ++i) {
      float4 x = rp4[i];
      pv[i].x = expf(x.x); pv[i].y = expf(x.y);
      pv[i].z = expf(x.z); pv[i].w = expf(x.w);
    }
    float rs = 0.f, s1 = 0.f;
    #pragma unroll
    for (int i = 0; i < 32; ++i) {
      rs += (pv[i].x + pv[i].y) + (pv[i].z + pv[i].w);
      s1 -= pv[i].x * logf(pv[i].x + EPS_) + pv[i].y * logf(pv[i].y + EPS_)
          + pv[i].z * logf(pv[i].z + EPS_) + pv[i].w * logf(pv[i].w + EPS_);
      rp4[i] = pv[i];
    }
    const float irs = 1.0f / rs;
    float s2e = 0.f;
    #pragma unroll
    for (int i = 0; i < 32; ++i) {
      float a = pv[i].x * irs, b = pv[i].y * irs;
      float c = pv[i].z * irs, d = pv[i].w * irs;
      s2e -= a * logf(a + EPS_) + b * logf(b + EPS_)
           + c * logf(c + EPS_) + d * logf(d + EPS_);
    }
    red[tid] = s1 + s2e;
  }
  __syncthreads();
  for (int off = 64; off > 0; off >>= 1) {
    if (tid < off) red[tid] += red[tid + off];
    __syncthreads();
  }
  if (tid == 0) ws_partial[k] = red[0];

  // ---- async store pw: LDS (pitch 132) -> global (pitch 128) --------------
  #pragma unroll
  for (int i = 0; i < 32; ++i) {
    int v  = tid + i * 128;
    int r  = v >> 5;
    int c4 = (v & 31) * 4;
    async_store_b128(dstPw + (r * E + c4), lbase + (uint32_t)(r * PITCH + c4) * 4u);
  }
  wait_async0();
}

// Fold 16 per-matrix entropy partials into the output scalar.
__global__ void finalize_reg_kernel(const float* __restrict__ ws_partial,
                                    float* __restrict__ out_scalar) {
  if (threadIdx.x == 0) {
    float s = 0.f;
    #pragma unroll
    for (int i = 0; i < KPERM; ++i) s += ws_partial[i];
    out_scalar[0] = 0.01f * (s / (float)KPERM);
  }
}

extern "C" void kernel_launch(void* const* d_in, const int* in_sizes, int n_in,
                              void* d_out, int out_size, void* d_ws, size_t ws_size,
                              hipStream_t stream) {
  (void)in_sizes; (void)n_in; (void)out_size; (void)ws_size;
  // d_in[0] = inputs [4096,768] (unused by the reference math)
  const float* plw = (const float*)d_in[1];   // [16,128,128]
  float* out = (float*)d_out;                 // 16*128*128 pw + 1 scalar
  float* ws  = (float*)d_ws;                  // 16 block partials

  // tau = 10^( log10(1e-3) + (log10(1e-7)-log10(1e-3)) * 1/5000 ) = 10^-3.0008
  const double log_tau = -3.0 + (-7.0 - (-3.0)) * (1.0 / 5000.0);
  const float inv_tau = (float)pow(10.0, -log_tau);

  const size_t shmem = (size_t)(PITCH * E + E + E) * sizeof(float);  // ~68.6 KB
  sinkhorn_entropy_kernel<<<KPERM, 128, shmem, stream>>>(plw, out, ws, inv_tau);
  finalize_reg_kernel<<<1, 32, 0, stream>>>(ws, out + (size_t)KPERM * E * E);
}